// TrajectoryRollout_73753178407522
// MI455X (gfx1250) — compile-verified
//
#include <hip/hip_runtime.h>

#define TOTAL_STEPS 1500
#define EVAL_START  500
#define BATCH       8192
#define HID         128

typedef __attribute__((ext_vector_type(2))) float v2f;
typedef __attribute__((ext_vector_type(8))) float v8f;

// Native V_TANH_F32 on gfx1250; branch-free exp2 fallback otherwise.
__device__ __forceinline__ float fast_tanh(float x) {
#if __has_builtin(__builtin_amdgcn_tanhf)
  return __builtin_amdgcn_tanhf(x);
#else
  float e = __builtin_amdgcn_exp2f(x * 2.8853900817779268f);  // exp(2x)
  return fmaf(-2.0f, __builtin_amdgcn_rcpf(e + 1.0f), 1.0f);
#endif
}

// Fold-reduce 8 per-lane partials over each 16-lane half.
// After the 4 butterfly stages, lane l holds the half-sum of variable (l&7),
// i.e. motor[8*(l>>4) + (l&7)] for this wave's 16-agent tile.
__device__ __forceinline__ float fold_reduce8(const float P[8], int lane) {
  const int j = lane & 15;
  float a[4], b[2], c;
#pragma unroll
  for (int i = 0; i < 4; ++i) {
    const bool hi = (j & 1) != 0;
    float keep = hi ? P[2 * i + 1] : P[2 * i];
    float send = hi ? P[2 * i]     : P[2 * i + 1];
    a[i] = keep + __shfl_xor(send, 1, 32);
  }
#pragma unroll
  for (int i = 0; i < 2; ++i) {
    const bool hi = (j & 2) != 0;
    float keep = hi ? a[2 * i + 1] : a[2 * i];
    float send = hi ? a[2 * i]     : a[2 * i + 1];
    b[i] = keep + __shfl_xor(send, 2, 32);
  }
  {
    const bool hi = (j & 4) != 0;
    float keep = hi ? b[1] : b[0];
    float send = hi ? b[0] : b[1];
    c = keep + __shfl_xor(send, 4, 32);
  }
  c += __shfl_xor(c, 8, 32);
  return c;
}

// One wave = 16 agents (state replicated in both 16-lane halves).
// Layer 1: V_WMMA_F32_16X16X4_F32, A = [16 agents x (ex,ey,s,1)],
//          B tile t = rows {W1[0,n], W1[1,n], W1[16,n], b1[n]}, n = t*16 + (lane&15).
// Layer 2: scalar-fma partials (dual-issued) + folding-butterfly reduction.
__global__ __launch_bounds__(256) void TrajectoryRollout_kernel(
    const float* __restrict__ all_food,   // [1500, B, 2]
    const float* __restrict__ agent_pos,  // [B, 2]
    const float* __restrict__ agent_vel,  // [B, 2]
    const float* __restrict__ noise_all,  // [1500, B, 1]
    const float* __restrict__ W1,         // [17, 128]
    const float* __restrict__ b1,         // [128]
    const float* __restrict__ W2,         // [128, 6]
    const float* __restrict__ b2,         // [6]
    float* __restrict__ out)              // [1000, B]
{
  const int lane   = threadIdx.x & 31;
  const int wave   = threadIdx.x >> 5;
  const int m      = lane & 15;                 // agent-in-tile == N index
  const bool lowHalf = (lane < 16);
  const int agent  = blockIdx.x * 128 + wave * 16 + m;

  // ---- Constant operands, resident in VGPRs for the whole rollout ----
  v2f   Bt[8];
  float w4[8], w5[8];
#pragma unroll
  for (int t = 0; t < 8; ++t) {
    const int n = t * 16 + m;
    // B 4x16 tile (layout mirrors documented A 16x4 layout):
    //   VGPR0: lanes0-15 -> K=0 row, lanes16-31 -> K=2 row
    //   VGPR1: lanes0-15 -> K=1 row, lanes16-31 -> K=3 row
    float e0 = lowHalf ? W1[0 * HID + n] : W1[16 * HID + n];
    float e1 = lowHalf ? W1[1 * HID + n] : b1[n];
    Bt[t][0] = e0;
    Bt[t][1] = e1;
    w4[t] = W2[n * 6 + 4];
    w5[t] = W2[n * 6 + 5];
  }
  const float bb4 = b2[4];
  const float bb5 = b2[5];

  float px = agent_pos[agent * 2 + 0];
  float py = agent_pos[agent * 2 + 1];
  float vx = agent_vel[agent * 2 + 0];
  float vy = agent_vel[agent * 2 + 1];

  for (int t = 0; t < TOTAL_STEPS; ++t) {
    const float* fp = all_food + ((size_t)t * BATCH + agent) * 2;
    float fx = fp[0];
    float fy = fp[1];
    float nz = noise_all[(size_t)t * BATCH + agent];

    // branchless stream-ahead into cache (global_prefetch_b8)
    {
      int tp = t + 4;
      tp = tp < TOTAL_STEPS ? tp : (TOTAL_STEPS - 1);
      __builtin_prefetch(all_food + ((size_t)tp * BATCH + agent) * 2, 0, 0);
      __builtin_prefetch(noise_all + (size_t)tp * BATCH + agent, 0, 0);
    }

    float dx   = fx - px;
    float dy   = fy - py;
    float dist = __builtin_amdgcn_sqrtf(fmaf(dx, dx, dy * dy));
    float invd = __builtin_amdgcn_rcpf(dist + 1e-5f);
    float ex   = dx * invd;
    float ey   = dy * invd;
    float st   = nz * (dist * 1e-5f);

    // A 16x4: lanes0-15 carry (K0,K1)=(ex,ey); lanes16-31 carry (K2,K3)=(s,1)
    v2f A;
    A[0] = lowHalf ? ex : st;
    A[1] = lowHalf ? ey : 1.0f;

    v8f h[8];
#pragma unroll
    for (int tt = 0; tt < 8; ++tt) {
      v8f c = {0.f, 0.f, 0.f, 0.f, 0.f, 0.f, 0.f, 0.f};
      h[tt] = __builtin_amdgcn_wmma_f32_16x16x4_f32(
          /*neg_a=*/false, A, /*neg_b=*/false, Bt[tt],
          /*c_mod=*/(short)0, c, /*reuse_a=*/false, /*reuse_b=*/false);
    }

#pragma unroll
    for (int tt = 0; tt < 8; ++tt)
#pragma unroll
      for (int r = 0; r < 8; ++r)
        h[tt][r] = fast_tanh(h[tt][r]);

    // ---- Layer 2 partials: Px[r], Py[r] = sum_t h_t[r] * W2col[t] ----
    float Px[8], Py[8];
#pragma unroll
    for (int r = 0; r < 8; ++r) {
      float sx = 0.f, sy = 0.f;
#pragma unroll
      for (int tt = 0; tt < 8; ++tt) {
        sx = fmaf(h[tt][r], w4[tt], sx);
        sy = fmaf(h[tt][r], w5[tt], sy);
      }
      Px[r] = sx;
      Py[r] = sy;
    }

    // Folding butterfly: lane l ends with motor[8*(l>>4) + (l&7)]
    float mx = fold_reduce8(Px, lane);
    float my = fold_reduce8(Py, lane);

    // agent (lane&15) lives in half (lane>>3)&1; swap halves if needed
    float ox = __shfl_xor(mx, 16, 32);
    float oy = __shfl_xor(my, 16, 32);
    const bool same = (((lane >> 3) & 1) == (lane >> 4));
    mx = (same ? mx : ox) + bb4;
    my = (same ? my : oy) + bb5;

    vx = vx * 0.85f + mx * 0.2f;
    vy = vy * 0.85f + my * 0.2f;
    px = fminf(fmaxf(px + vx, -4.0f), 4.0f);
    py = fminf(fmaxf(py + vy, -4.0f), 4.0f);

    if (lowHalf && t >= EVAL_START)
      out[(size_t)(t - EVAL_START) * BATCH + agent] = dist;
  }
}

extern "C" void kernel_launch(void* const* d_in, const int* in_sizes, int n_in,
                              void* d_out, int out_size, void* d_ws, size_t ws_size,
                              hipStream_t stream) {
  const float* all_food  = (const float*)d_in[0];
  const float* agent_pos = (const float*)d_in[1];
  const float* agent_vel = (const float*)d_in[2];
  const float* noise_all = (const float*)d_in[3];
  const float* W1        = (const float*)d_in[4];
  const float* b1        = (const float*)d_in[5];
  const float* W2        = (const float*)d_in[6];
  const float* b2        = (const float*)d_in[7];
  float* out             = (float*)d_out;

  // 8192 agents, 16 per wave, 8 waves (128 agents) per block -> 64 blocks
  dim3 grid(BATCH / 128);
  dim3 block(256);
  TrajectoryRollout_kernel<<<grid, block, 0, stream>>>(
      all_food, agent_pos, agent_vel, noise_all, W1, b1, W2, b2, out);
}